// GraphConvNet_30966714204196
// MI455X (gfx1250) — compile-verified
//
#include <hip/hip_runtime.h>
#include <cstdint>
#include <cstddef>

#define MAX_DEG 10
#define NNODE   512
#define BATCH   128
#define FDIM    75
#define HDIM    128
#define TDIM    12
#define TOTAL_ROWS (BATCH * NNODE)   // 65536
#define EPSV    1e-5f

typedef __attribute__((ext_vector_type(2))) float v2f;
typedef __attribute__((ext_vector_type(8))) float v8f;

// ---------------------------------------------------------------------------
// Graph conv layer: out[m,:] = x[m,:] @ W_self[deg[m]] + b_self[deg[m]]
//                            + (deg>0 ? ns[m,:] @ W_neigh[deg-1] + b_neigh[deg-1] : 0)
// ns computed in-kernel (fused gather). Degree selection folded into the A
// operand as a per-row {0,1} mask so all 21 degree passes accumulate into a
// single v_wmma_f32_16x16x4_f32 accumulator chain.
// Workgroup: 256 threads / 8 waves, tile = 128 rows x 128 cols.
// ---------------------------------------------------------------------------
template <int K_IN, int K_PAD>
__global__ __launch_bounds__(256) void graphconv_wmma(
    const float* __restrict__ src,      // [TOTAL_ROWS, K_IN]
    const int*   __restrict__ adj,      // [TOTAL_ROWS, MAX_DEG]
    const int*   __restrict__ deg,      // [TOTAL_ROWS]
    const float* __restrict__ W_self,   // [MAX_DEG+1, K_IN, HDIM]
    const float* __restrict__ W_neigh,  // [MAX_DEG,   K_IN, HDIM]
    const float* __restrict__ b_self,   // [MAX_DEG+1, HDIM]
    const float* __restrict__ b_neigh,  // [MAX_DEG,   HDIM]
    float* __restrict__ out)            // [TOTAL_ROWS, HDIM]
{
    __shared__ float xs[128 * K_PAD];
    __shared__ float ns[128 * K_PAD];
    __shared__ int   adj_s[128 * MAX_DEG];
    __shared__ int   deg_s[128];

    const int tid   = threadIdx.x;
    const int m0    = blockIdx.x * 128;
    const int bbase = (m0 / NNODE) * NNODE;   // batch-aligned (512 % 128 == 0)

    for (int i = tid; i < 128; i += 256) deg_s[i] = deg[m0 + i];
    for (int i = tid; i < 128 * MAX_DEG; i += 256)
        adj_s[i] = adj[(size_t)m0 * MAX_DEG + i];
    for (int i = tid; i < 128 * K_PAD; i += 256) {
        int r = i / K_PAD, f = i - r * K_PAD;
        xs[i] = (f < K_IN) ? src[(size_t)(m0 + r) * K_IN + f] : 0.f;
    }
    __syncthreads();

    // fused neighbor-sum into LDS (gathers stay within one batch's 512-row block)
    const float* srcb = src + (size_t)bbase * K_IN;
    for (int i = tid; i < 128 * K_PAD; i += 256) {
        int r = i / K_PAD, f = i - r * K_PAD;
        float s = 0.f;
        if (f < K_IN) {
            int dr = deg_s[r];
            for (int k = 0; k < dr; ++k)
                s += srcb[(size_t)adj_s[r * MAX_DEG + k] * K_IN + f];
        }
        ns[i] = s;
    }
    __syncthreads();

    const int lane  = tid & 31;
    const int wv    = tid >> 5;          // wave id 0..7 -> column tile
    const int col   = lane & 15;         // B/C/D column within tile
    const int rowA  = lane & 15;         // A row within 16-row subtile
    const int khalf = (lane >> 4) << 1;  // lanes 16..31 hold K+2,K+3
    const int c0    = wv * 16;

    v8f acc[8] = {};

    for (int d = 0; d <= MAX_DEG; ++d) {
        float msk[8];
#pragma unroll
        for (int s = 0; s < 8; ++s)
            msk[s] = (deg_s[s * 16 + rowA] == d) ? 1.f : 0.f;

        // ---- self path: (mask_d * X) @ W_self[d] ----
        {
            const float* Wd = W_self + (size_t)d * K_IN * HDIM;
            for (int ks = 0; ks < K_PAD / 4; ++ks) {
                const int k  = ks * 4 + khalf;
                const int k0 = (k     < K_IN) ? k     : (K_IN - 1);
                const int k1 = (k + 1 < K_IN) ? k + 1 : (K_IN - 1);
                float b0v = Wd[(size_t)k0 * HDIM + c0 + col];
                float b1v = Wd[(size_t)k1 * HDIM + c0 + col];
                v2f bf;
                bf.x = (k     < K_IN) ? b0v : 0.f;
                bf.y = (k + 1 < K_IN) ? b1v : 0.f;
#pragma unroll
                for (int s = 0; s < 8; ++s) {
                    const float* ap = &xs[(s * 16 + rowA) * K_PAD + k];
                    v2f af; af.x = ap[0] * msk[s]; af.y = ap[1] * msk[s];
                    acc[s] = __builtin_amdgcn_wmma_f32_16x16x4_f32(
                        false, af, false, bf, (short)0, acc[s], false, false);
                }
            }
        }
        // ---- neighbor path: (mask_d * NS) @ W_neigh[d-1], d >= 1 ----
        if (d >= 1) {
            const float* Wd = W_neigh + (size_t)(d - 1) * K_IN * HDIM;
            for (int ks = 0; ks < K_PAD / 4; ++ks) {
                const int k  = ks * 4 + khalf;
                const int k0 = (k     < K_IN) ? k     : (K_IN - 1);
                const int k1 = (k + 1 < K_IN) ? k + 1 : (K_IN - 1);
                float b0v = Wd[(size_t)k0 * HDIM + c0 + col];
                float b1v = Wd[(size_t)k1 * HDIM + c0 + col];
                v2f bf;
                bf.x = (k     < K_IN) ? b0v : 0.f;
                bf.y = (k + 1 < K_IN) ? b1v : 0.f;
#pragma unroll
                for (int s = 0; s < 8; ++s) {
                    const float* ap = &ns[(s * 16 + rowA) * K_PAD + k];
                    v2f af; af.x = ap[0] * msk[s]; af.y = ap[1] * msk[s];
                    acc[s] = __builtin_amdgcn_wmma_f32_16x16x4_f32(
                        false, af, false, bf, (short)0, acc[s], false, false);
                }
            }
        }
    }

    // bias (degree-selected) + store. C/D layout: lane holds row v + 8*(lane>=16), col lane&15.
    const int rhi = (lane >> 4) << 3;
#pragma unroll
    for (int s = 0; s < 8; ++s) {
#pragma unroll
        for (int v = 0; v < 8; ++v) {
            const int rloc = s * 16 + v + rhi;
            const int dr   = deg_s[rloc];
            const int c    = c0 + col;
            float bias = b_self[dr * HDIM + c];
            const int dn = (dr > 0) ? (dr - 1) : 0;
            float bnv = b_neigh[dn * HDIM + c];
            bias += (dr > 0) ? bnv : 0.f;
            out[(size_t)(m0 + rloc) * HDIM + c] = acc[s][v] + bias;
        }
    }
}

// ---------------------------------------------------------------------------
// Per-feature sum / sum-of-squares over all 65536 rows (for BatchNorm).
// stats[0..127] = sum, stats[128..255] = sumsq. Grid 256 x 128 threads.
// ---------------------------------------------------------------------------
__global__ __launch_bounds__(128) void bn_stats_kernel(
    const float* __restrict__ h, float* __restrict__ stats)
{
    const int c  = threadIdx.x;            // feature column
    const int r0 = blockIdx.x * 256;
    float s = 0.f, q = 0.f;
    for (int r = 0; r < 256; ++r) {
        float v = h[(size_t)(r0 + r) * HDIM + c];
        s += v; q += v * v;
    }
    atomicAdd(&stats[c], s);
    atomicAdd(&stats[HDIM + c], q);
}

__global__ __launch_bounds__(512) void zero_stats_kernel(float* __restrict__ stats)
{
    if (threadIdx.x < 512) stats[threadIdx.x] = 0.f;
}

// ---------------------------------------------------------------------------
// BatchNorm + ReLU elementwise: out = relu((h - mu) * rsqrt(var+eps) * g + b)
// ---------------------------------------------------------------------------
__global__ __launch_bounds__(256) void bn_relu_kernel(
    const float* __restrict__ h, const float* __restrict__ stats,
    const float* __restrict__ gamma, const float* __restrict__ beta,
    float* __restrict__ out)
{
    const size_t i = (size_t)blockIdx.x * 256 + threadIdx.x;
    const int c = (int)(i & (HDIM - 1));
    const float inv = 1.f / (float)TOTAL_ROWS;
    float mu  = stats[c] * inv;
    float var = stats[HDIM + c] * inv - mu * mu;
    float v = (h[i] - mu) * rsqrtf(var + EPSV) * gamma[c] + beta[c];
    out[i] = v > 0.f ? v : 0.f;
}

// ---------------------------------------------------------------------------
// Fused: BN2 + ReLU + mean over N + dense (128 -> 12). One block per batch b.
// ---------------------------------------------------------------------------
__global__ __launch_bounds__(128) void final_kernel(
    const float* __restrict__ h2, const float* __restrict__ stats2,
    const float* __restrict__ gamma, const float* __restrict__ beta,
    const float* __restrict__ Wd, const float* __restrict__ bd,
    float* __restrict__ out)
{
    __shared__ float pooled[HDIM];
    const int b = blockIdx.x;
    const int t = threadIdx.x;     // feature index
    const float inv = 1.f / (float)TOTAL_ROWS;
    float mu  = stats2[t] * inv;
    float var = stats2[HDIM + t] * inv - mu * mu;
    float sc  = rsqrtf(var + EPSV) * gamma[t];
    float sh  = beta[t];

    const float* base = h2 + (size_t)b * NNODE * HDIM;
    float acc = 0.f;
    for (int n = 0; n < NNODE; ++n) {
        float v = (base[(size_t)n * HDIM + t] - mu) * sc + sh;
        acc += v > 0.f ? v : 0.f;
    }
    pooled[t] = acc * (1.f / (float)NNODE);
    __syncthreads();

    if (t < TDIM) {
        float s = bd[t];
        for (int hh = 0; hh < HDIM; ++hh)
            s += pooled[hh] * Wd[hh * TDIM + t];
        out[b * TDIM + t] = s;
    }
}

// ---------------------------------------------------------------------------
extern "C" void kernel_launch(void* const* d_in, const int* in_sizes, int n_in,
                              void* d_out, int out_size, void* d_ws, size_t ws_size,
                              hipStream_t stream)
{
    (void)in_sizes; (void)n_in; (void)out_size; (void)ws_size;
    const float* x        = (const float*)d_in[0];
    const int*   adj      = (const int*)  d_in[1];
    const int*   deg      = (const int*)  d_in[2];
    const float* W_neigh1 = (const float*)d_in[3];
    const float* W_self1  = (const float*)d_in[4];
    const float* b_neigh1 = (const float*)d_in[5];
    const float* b_self1  = (const float*)d_in[6];
    const float* gamma1   = (const float*)d_in[7];
    const float* beta1    = (const float*)d_in[8];
    const float* W_neigh2 = (const float*)d_in[9];
    const float* W_self2  = (const float*)d_in[10];
    const float* b_neigh2 = (const float*)d_in[11];
    const float* b_self2  = (const float*)d_in[12];
    const float* gamma2   = (const float*)d_in[13];
    const float* beta2    = (const float*)d_in[14];
    const float* W_dense  = (const float*)d_in[15];
    const float* b_dense  = (const float*)d_in[16];
    float*       out      = (float*)d_out;

    // workspace: buf0 = h1 (later reused as h2), buf1 = h1n, then stats (512 f32)
    const size_t HBYTES = (size_t)TOTAL_ROWS * HDIM * sizeof(float);  // 33.5 MB
    uint8_t* ws = (uint8_t*)d_ws;
    float* h1     = (float*)(ws);                 // layer-1 pre-BN; reused for h2
    float* h1n    = (float*)(ws + HBYTES);        // layer-1 post BN+ReLU
    float* stats  = (float*)(ws + 2 * HBYTES);    // [0..255]=BN1, [256..511]=BN2
    float* stats1 = stats;
    float* stats2 = stats + 256;
    float* h2     = h1;

    zero_stats_kernel<<<1, 512, 0, stream>>>(stats);

    // layer 1: F=75 -> H=128 (K padded to 76)
    graphconv_wmma<FDIM, 76><<<TOTAL_ROWS / 128, 256, 0, stream>>>(
        x, adj, deg, W_self1, W_neigh1, b_self1, b_neigh1, h1);

    bn_stats_kernel<<<TOTAL_ROWS / 256, 128, 0, stream>>>(h1, stats1);
    bn_relu_kernel<<<(TOTAL_ROWS * HDIM) / 256, 256, 0, stream>>>(
        h1, stats1, gamma1, beta1, h1n);

    // layer 2: H=128 -> H=128 (K already multiple of 4); h2 overwrites h1
    graphconv_wmma<HDIM, HDIM><<<TOTAL_ROWS / 128, 256, 0, stream>>>(
        h1n, adj, deg, W_self2, W_neigh2, b_self2, b_neigh2, h2);

    bn_stats_kernel<<<TOTAL_ROWS / 256, 128, 0, stream>>>(h2, stats2);

    final_kernel<<<BATCH, HDIM, 0, stream>>>(
        h2, stats2, gamma2, beta2, W_dense, b_dense, out);
}